// VirtualParameter_85203561218152
// MI455X (gfx1250) — compile-verified
//
#include <hip/hip_runtime.h>

typedef __attribute__((ext_vector_type(2))) float v2f;
typedef __attribute__((ext_vector_type(8))) float v8f;

#define BANKS 64
#define NB    128            // selection contexts (B)
#define KSEL  4              // selected banks per context
#define D_IO  (1024 * 1024)  // D_IN * D_OUT flattened

// ---------------------------------------------------------------------------
// Step 1: densify the sparse selection into W[b, bank] (128 x 64 = 32 KB).
// W[b,bank] = sum_k probs[b,k] * (idx[b,k] == bank); duplicates accumulate,
// matching the einsum over k in the reference.
// ---------------------------------------------------------------------------
__global__ __launch_bounds__(256) void vp_build_w(
    const long long* __restrict__ idx,   // (128,4) int64
    const float* __restrict__ probs,     // (128,4) f32
    float* __restrict__ W)               // (128,64) f32, row-major
{
    int t = blockIdx.x * 256 + threadIdx.x;
    if (t >= NB * BANKS) return;
    int b = t >> 6;
    int bank = t & 63;
    float w = 0.f;
#pragma unroll
    for (int k = 0; k < KSEL; ++k) {
        long long sel = idx[b * KSEL + k];
        w += (sel == (long long)bank) ? probs[b * KSEL + k] : 0.f;
    }
    W[t] = w;
}

// ---------------------------------------------------------------------------
// Step 2: out[128 x 1M] = W[128 x 64] @ param^T[64 x 1M] with f32 WMMA.
//
// Wave-per-N-tile: each wave owns 16 io positions (rows of param), loads the
// 16x64 param slab once (16 K-tiles of v2f per lane), then accumulates all
// 8 M-tiles (16 b's each) with 16 chained V_WMMA_F32_16X16X4_F32 per tile.
// param is read exactly once from HBM; W (32 KB) stays cache-resident.
// ---------------------------------------------------------------------------
__global__ __launch_bounds__(256) void vp_wmma_gemm(
    const float* __restrict__ W,     // (128,64)
    const float* __restrict__ P,     // param viewed as (D_IO, 64), bank stride 1
    float* __restrict__ out)         // (128, D_IO)
{
    const int lane = threadIdx.x & 31;
    const int wave = threadIdx.x >> 5;
    const int half = lane >> 4;      // 0: lanes 0-15, 1: lanes 16-31
    const int m    = lane & 15;
    const long n0  = ((long)blockIdx.x * 8 + wave) * 16;   // io base of N-tile

    // B operand tiles: B[K=4kt+2*half (+1), N=m] = P[(n0+m)*64 + 4kt + 2*half]
    v2f Bt[16];
    const float* pb = P + (n0 + (long)m) * BANKS + 2 * half;
#pragma unroll
    for (int kt = 0; kt < 16; ++kt)
        Bt[kt] = *(const v2f*)(pb + 4 * kt);

    for (int mt = 0; mt < 8; ++mt) {
        // A operand: A[M=m, K=4kt+2*half (+1)] = W[(16mt+m)*64 + 4kt + 2*half]
        const float* wb = W + (16 * mt + m) * BANKS + 2 * half;
        v8f acc = {};
#pragma unroll
        for (int kt = 0; kt < 16; ++kt) {
            v2f At = *(const v2f*)(wb + 4 * kt);
            // 8 args: (neg_a, A, neg_b, B, c_mod, C, reuse_a, reuse_b)
            acc = __builtin_amdgcn_wmma_f32_16x16x4_f32(
                false, At, false, Bt[kt], (short)0, acc, false, false);
        }
        // D layout: VGPR r -> row (r + 8*half), col m
        float* ob = out + ((long)(16 * mt + 8 * half)) * D_IO + n0 + m;
#pragma unroll
        for (int r = 0; r < 8; ++r)
            ob[(long)r * D_IO] = acc[r];
    }
}

extern "C" void kernel_launch(void* const* d_in, const int* in_sizes, int n_in,
                              void* d_out, int out_size, void* d_ws, size_t ws_size,
                              hipStream_t stream) {
    const long long* idx   = (const long long*)d_in[0];  // int64 selection_index
    const float*     probs = (const float*)d_in[1];
    const float*     param = (const float*)d_in[2];
    float*           out   = (float*)d_out;
    float*           W     = (float*)d_ws;               // 32 KB scratch

    vp_build_w<<<(NB * BANKS + 255) / 256, 256, 0, stream>>>(idx, probs, W);

    // 1M io / (16 per wave * 8 waves per block) = 8192 blocks
    vp_wmma_gemm<<<D_IO / 128, 256, 0, stream>>>(W, param, out);
}